// GraphEncoder_62534723830351
// MI455X (gfx1250) — compile-verified
//
#include <hip/hip_runtime.h>
#include <stdint.h>

typedef float v2f __attribute__((ext_vector_type(2)));
typedef float v8f __attribute__((ext_vector_type(8)));
typedef unsigned int v4u __attribute__((ext_vector_type(4)));
typedef int v8i __attribute__((ext_vector_type(8)));
typedef int v4i __attribute__((ext_vector_type(4)));

#if __has_builtin(__builtin_amdgcn_tensor_load_to_lds) && \
    __has_builtin(__builtin_amdgcn_s_wait_tensorcnt)
#define USE_TDM 1
#else
#define USE_TDM 0
#endif

// ---------------------------------------------------------------------------
// Zero-fill (float4 wide). Sizes here are always multiples of 4 floats.
// ---------------------------------------------------------------------------
__global__ __launch_bounds__(256) void zero_f32(float4* __restrict__ p, long long n4) {
    long long i = (long long)blockIdx.x * blockDim.x + threadIdx.x;
    long long stride = (long long)gridDim.x * blockDim.x;
    float4 z; z.x = 0.f; z.y = 0.f; z.z = 0.f; z.w = 0.f;
    for (; i < n4; i += stride) p[i] = z;
}

#if USE_TDM
// ---------------------------------------------------------------------------
// TDM: DMA CHUNK rows x OUT cols of W (f32, row-major) into LDS at
// lds_byte_off, padding +16 DWORDs after each row (LDS stride OUT+16 floats,
// bank-conflict-free B-fragment reads). ISA 8.3/8.4 descriptor layout:
//   group0 = {count=1, lds_addr, global_addr[56:0], type=2}
//   group1 = {data_size=4B, pad_enable, pad_interval, pad_amount=16 DWORDs,
//             tensor_dim0=OUT, tensor_dim1=K, tile_dim0=OUT, tile_dim1=CHUNK,
//             tensor_dim0_stride=OUT}
// Groups 2/3 disabled (2-D tile). This toolchain's builtin is the 6-arg
// clang-23 form (extra int32x8 operand before cpol) — zero-filled.
// PAD_IV: pad_interval with 2^(v+1) DWORDs per row: OUT=256 -> 7, 128 -> 6.
// ---------------------------------------------------------------------------
template <int OUT, int K, int CHUNK, int PAD_IV>
__device__ __forceinline__ void tdm_load_rows(const float* Wg, unsigned lds_byte_off,
                                              int row0) {
    uint64_t ga = (uint64_t)(uintptr_t)(Wg + (size_t)row0 * OUT);
    v4u g0;
    g0.x = 1u;                                                 // count=1 user D#
    g0.y = lds_byte_off;                                       // lds_addr
    g0.z = (unsigned)(ga & 0xffffffffu);                       // global_addr[31:0]
    g0.w = (unsigned)((ga >> 32) & 0x01ffffffu) | (2u << 30);  // addr[56:32] | type=2
    v8i g1;
    // w0: wg_mask=0 | data_size=2(4B)<<16 | pad_enable<<20 | pad_iv<<22 | pad_amt<<25
    g1[0] = (int)((2u << 16) | (1u << 20) | ((unsigned)PAD_IV << 22) | (15u << 25));
    g1[1] = (int)(((unsigned)OUT & 0xffffu) << 16);            // tensor_dim0 lo16
    g1[2] = (int)((((unsigned)K & 0xffffu) << 16) |            // tensor_dim1 lo16
                  (((unsigned)OUT >> 16) & 0xffffu));          // tensor_dim0 hi16
    g1[3] = (int)((((unsigned)OUT & 0xffffu) << 16) |          // tile_dim0 = OUT
                  (((unsigned)K >> 16) & 0xffffu));            // tensor_dim1 hi16
    g1[4] = CHUNK;                                             // tile_dim1 rows, tile_dim2=0
    g1[5] = OUT;                                               // tensor_dim0_stride lo32
    g1[6] = 0;
    g1[7] = 0;
    v4i gz4 = (v4i){0, 0, 0, 0};                               // groups 2/3 off (2-D)
    v8i gz8 = (v8i){0, 0, 0, 0, 0, 0, 0, 0};                   // trailing operand (clang-23)
    __builtin_amdgcn_tensor_load_to_lds(g0, g1, gz4, gz4, gz8, 0);
}
#endif

// ---------------------------------------------------------------------------
// GEMM + bias (+ optional fused ReLU on the *input*), f32 WMMA 16x16x4.
//   H[m,n] = sum_k act(X[m,k]) * W[k,n] + bias[n]
// Workgroup = 4 waves, one 16-row strip of X per block staged in LDS
// (row stride K+4 floats: bank-conflict-free + 16B aligned). Each wave owns
// TILES 16x16 output tiles. W is streamed k-chunk by k-chunk into a
// double-buffered LDS tile via the Tensor Data Mover (wave 0 issues DMA for
// chunk c+1 while all waves compute chunk c; s_wait_tensorcnt + barrier).
// A-frag layout (ISA 7.12.2, 32-bit A 16x4): v0 = K{0 | 2}, v1 = K{1 | 3}
// split across lane halves; B-frag symmetric (rows striped across lanes).
// ---------------------------------------------------------------------------
template <int K, int OUT, int TILES, bool RELU>
__global__ __launch_bounds__(128) void gemm_bias_wmma(
        const float* __restrict__ X, const float* __restrict__ W,
        const float* __restrict__ bias, float* __restrict__ H, int n_rows) {
    constexpr int XS = K + 4;                   // 260 for K=256
    __shared__ float xs[16 * XS];

    const int lane = threadIdx.x & 31;
    const int wave = threadIdx.x >> 5;
    const int half = lane >> 4;                 // 0: lanes 0-15, 1: lanes 16-31
    const int l16  = lane & 15;
    const int m0   = blockIdx.x * 16;
    const int n_base = wave * (TILES * 16);

    // Cooperative, coalesced float4 staging of the 16 x K X-strip into LDS.
    for (int i = threadIdx.x; i < 16 * (K / 4); i += blockDim.x) {
        int r  = i / (K / 4);
        int c4 = i % (K / 4);
        int rr = m0 + r; if (rr >= n_rows) rr = n_rows - 1;   // defensive clamp
        float4 v = ((const float4*)(X + (size_t)rr * K))[c4];
        if (RELU) {
            v.x = fmaxf(v.x, 0.f); v.y = fmaxf(v.y, 0.f);
            v.z = fmaxf(v.z, 0.f); v.w = fmaxf(v.w, 0.f);
        }
        *(float4*)(&xs[r * XS + c4 * 4]) = v;
    }

    v8f acc[TILES];
#pragma unroll
    for (int t = 0; t < TILES; ++t) acc[t] = (v8f){0.f,0.f,0.f,0.f,0.f,0.f,0.f,0.f};

#if USE_TDM
    constexpr int CHUNK  = 16;                  // k-rows of W per DMA
    constexpr int WS     = OUT + 16;            // padded LDS row stride (floats)
    constexpr int PAD_IV = (OUT == 256) ? 7 : 6;
    constexpr int NCHUNK = K / CHUNK;
    __shared__ float wsh[2][CHUNK * WS];

    if (wave == 0) {
        tdm_load_rows<OUT, K, CHUNK, PAD_IV>(
            W, (unsigned)(uintptr_t)(&wsh[0][0]), 0);
        __builtin_amdgcn_s_wait_tensorcnt(0);
    }
    __syncthreads();                            // X strip + W chunk 0 visible

    for (int c = 0; c < NCHUNK; ++c) {
        const int buf = c & 1;
        if (wave == 0 && (c + 1) < NCHUNK) {    // prefetch next chunk via TDM
            tdm_load_rows<OUT, K, CHUNK, PAD_IV>(
                W, (unsigned)(uintptr_t)(&wsh[buf ^ 1][0]), (c + 1) * CHUNK);
        }
        const float* wb = &wsh[buf][0];
#pragma unroll
        for (int kk = 0; kk < CHUNK; kk += 4) {
            const int k0 = c * CHUNK + kk;
            v2f a;
            a.x = xs[l16 * XS + k0 + 2 * half];
            a.y = xs[l16 * XS + k0 + 2 * half + 1];
#pragma unroll
            for (int t = 0; t < TILES; ++t) {
                const float* wp = wb + (kk + 2 * half) * WS + n_base + t * 16 + l16;
                v2f b;
                b.x = wp[0];                    // K = k0 + 2*half
                b.y = wp[WS];                   // K = k0 + 2*half + 1
                acc[t] = __builtin_amdgcn_wmma_f32_16x16x4_f32(
                    false, a, false, b, (short)0, acc[t], false, false);
            }
        }
        if (wave == 0) __builtin_amdgcn_s_wait_tensorcnt(0);
        __syncthreads();                        // next buffer landed, safe to swap
    }
#else
    // Fallback (no TDM builtin / host parse): B fragments straight from global.
    __syncthreads();
#pragma unroll 4
    for (int k0 = 0; k0 < K; k0 += 4) {
        v2f a;
        a.x = xs[l16 * XS + k0 + 2 * half];
        a.y = xs[l16 * XS + k0 + 2 * half + 1];
#pragma unroll
        for (int t = 0; t < TILES; ++t) {
            const float* wp = W + (size_t)(k0 + 2 * half) * OUT + n_base + t * 16 + l16;
            v2f b;
            b.x = wp[0];
            b.y = wp[OUT];
            acc[t] = __builtin_amdgcn_wmma_f32_16x16x4_f32(
                false, a, false, b, (short)0, acc[t], false, false);
        }
    }
#endif

    // Epilogue: fused bias, store. C layout: VGPR r -> M = m0 + r + 8*half,
    // lane%16 -> N = n_base + t*16 + l16.
#pragma unroll
    for (int t = 0; t < TILES; ++t) {
        const int n  = n_base + t * 16 + l16;
        const float bv = bias[n];
#pragma unroll
        for (int r = 0; r < 8; ++r) {
            const int m = m0 + r + 8 * half;
            if (m < n_rows) H[(size_t)m * OUT + n] = acc[t][r] + bv;
        }
    }
}

// ---------------------------------------------------------------------------
// Edge scatter: out[dst[e]] += w[e] * H[src[e]]  (row of D floats per edge).
// Thread = (edge, float4 chunk): wide coalesced gather + 4 f32 atomics.
// ---------------------------------------------------------------------------
template <int D>
__global__ __launch_bounds__(256) void scatter_edges(
        const float* __restrict__ H, const int* __restrict__ src,
        const int* __restrict__ dst, const float* __restrict__ ew,
        float* __restrict__ out, int nE) {
    constexpr int CH = D / 4;
    long long t = (long long)blockIdx.x * blockDim.x + threadIdx.x;
    const long long total = (long long)nE * CH;
    if (t >= total) return;
    const int e = (int)(t / CH);
    const int j = (int)(t % CH);
    const int u = src[e];
    const int v = dst[e];
    const float w = ew[e];
    const float4 hv = ((const float4*)(H + (size_t)u * D))[j];
    float* o = out + (size_t)v * D + (size_t)j * 4;
    atomicAdd(o + 0, hv.x * w);
    atomicAdd(o + 1, hv.y * w);
    atomicAdd(o + 2, hv.z * w);
    atomicAdd(o + 3, hv.w * w);
}

// ---------------------------------------------------------------------------
// Launch: zero accumulators -> GEMM1 -> scatter1 -> GEMM2(relu-in) -> scatter2
// ---------------------------------------------------------------------------
extern "C" void kernel_launch(void* const* d_in, const int* in_sizes, int n_in,
                              void* d_out, int out_size, void* d_ws, size_t ws_size,
                              hipStream_t stream) {
    constexpr int IN_DIM = 256, HID = 256, EMB = 128;

    const float* x   = (const float*)d_in[0];
    const int*   ei  = (const int*)d_in[1];
    const float* ew  = (const float*)d_in[2];
    const float* W1  = (const float*)d_in[3];
    const float* b1  = (const float*)d_in[4];
    const float* W2  = (const float*)d_in[5];
    const float* b2  = (const float*)d_in[6];
    float*       out = (float*)d_out;

    const int N = in_sizes[0] / IN_DIM;      // 100000
    const int E = in_sizes[2];               // 1600000
    const int* src = ei;                     // edge_index[0, :]
    const int* dst = ei + E;                 // edge_index[1, :]

    // Workspace layout (floats): H1[N*HID] | A1[N*HID] | H2[N*EMB]
    float* H1 = (float*)d_ws;
    float* A1 = H1 + (size_t)N * HID;
    float* H2 = A1 + (size_t)N * HID;

    // Zero aggregation targets (every call: deterministic under graph replay).
    {
        long long n4a = ((long long)N * HID) / 4;
        long long n4o = ((long long)N * EMB) / 4;
        zero_f32<<<4096, 256, 0, stream>>>((float4*)A1, n4a);
        zero_f32<<<4096, 256, 0, stream>>>((float4*)out, n4o);
    }

    const int mblocks = (N + 15) / 16;       // 6250

    // Layer 1 linear: H1 = x @ W1 + b1
    gemm_bias_wmma<IN_DIM, HID, 4, false><<<mblocks, 128, 0, stream>>>(x, W1, b1, H1, N);

    // Layer 1 aggregate: A1[v] += w_e * H1[u]
    {
        long long total = (long long)E * (HID / 4);
        unsigned blocks = (unsigned)((total + 255) / 256);
        scatter_edges<HID><<<blocks, 256, 0, stream>>>(H1, src, dst, ew, A1, E);
    }

    // Layer 2 linear with fused ReLU on input: H2 = relu(A1) @ W2 + b2
    gemm_bias_wmma<HID, EMB, 2, true><<<mblocks, 128, 0, stream>>>(A1, W2, b2, H2, N);

    // Layer 2 aggregate: out[v] += w_e * H2[u]
    {
        long long total = (long long)E * (EMB / 4);
        unsigned blocks = (unsigned)((total + 255) / 256);
        scatter_edges<EMB><<<blocks, 256, 0, stream>>>(H2, src, dst, ew, out, E);
    }
}